// Yolov1_Loss_29746943492238
// MI455X (gfx1250) — compile-verified
//
#include <hip/hip_runtime.h>

typedef __attribute__((ext_vector_type(2))) float v2f;
typedef __attribute__((ext_vector_type(8))) float v8f;

#define YS 7
#define YB 2
#define YC 2
#define YCH (YB * 5 + YC)       // 12 channels per cell, both tensors
#define IMG_F 448.0f
#define GRID_F 64.0f
#define L_COORD 5.0f
#define L_NOOBJ 0.5f

// IOU of one predicted box vs the (precomputed) GT box, matching the
// reference's trunc/clip semantics exactly.
__device__ __forceinline__ float iou_box(float x, float y, float w, float h,
                                         float gx, float gy,
                                         float glx, float guy, float grx, float gdy,
                                         float g_area) {
    float px = truncf(gx + x * GRID_F);
    float py = truncf(gy + y * GRID_F);
    float pw = truncf(w * IMG_F);
    float ph = truncf(h * IMG_F);
    float plx = fmaxf(0.0f, px - pw * 0.5f);
    float puy = fmaxf(0.0f, py - ph * 0.5f);
    float prx = fminf(IMG_F - 1.0f, px + pw * 0.5f);
    float pdy = fminf(IMG_F - 1.0f, py + ph * 0.5f);
    float p_area = (prx - plx) * (pdy - puy);
    float cl = fmaxf(plx, glx);
    float cr = fminf(prx, grx);
    float cu = fmaxf(puy, guy);
    float cd = fminf(pdy, gdy);
    float inter = (cr - cl + 1.0f) * (cd - cu + 1.0f);
    bool sep = (cr < cl) || (cd < cu);
    return sep ? 0.0f : inter / (p_area + g_area - inter);
}

__global__ void yolo_zero_kernel(float* out) {
    if (threadIdx.x == 0 && blockIdx.x == 0) out[0] = 0.0f;
}

__global__ __launch_bounds__(256)
void yolo_loss_kernel(const float* __restrict__ bb,   // (N,7,7,12)
                      const float* __restrict__ gt,   // (N,7,7,1,12) -> same flat layout
                      float* __restrict__ out,
                      int cells) {
    const int tid    = blockIdx.x * blockDim.x + threadIdx.x;
    const int stride = gridDim.x * blockDim.x;

    float acc = 0.0f;

    for (int c = tid; c < cells; c += stride) {
        const float* pb = bb + (size_t)c * YCH;
        const float* pg = gt + (size_t)c * YCH;

        // Prefetch next iteration's lines (gfx1250 global_prefetch_b8).
        int cn = c + stride;
        if (cn < cells) {
            __builtin_prefetch(bb + (size_t)cn * YCH, 0, 0);
            __builtin_prefetch(gt + (size_t)cn * YCH, 0, 0);
        }

        // 12 floats = 3 x b128 per tensor; 48B stride keeps lines fully used.
        float4 b0 = *(const float4*)(pb + 0);   // x0 y0 w0 h0
        float4 b1 = *(const float4*)(pb + 4);   // c0 x1 y1 w1
        float4 b2 = *(const float4*)(pb + 8);   // h1 c1 pc0 pc1
        float4 g0 = *(const float4*)(pg + 0);   // gx gy gw gh
        float4 g1 = *(const float4*)(pg + 4);   // gconf glx guy grx
        float4 g2 = *(const float4*)(pg + 8);   // gdy exist gc0 gc1

        // Cell coordinates: flat = ((n*7 + i)*7 + j)
        int r = c % (YS * YS);
        int i = r / YS;
        int j = r % YS;
        float gxc = (float)i * GRID_F;
        float gyc = (float)j * GRID_F;

        // pred boxes
        float x0 = b0.x, y0 = b0.y, w0 = b0.z, h0 = b0.w, cf0 = b1.x;
        float x1 = b1.y, y1 = b1.z, w1 = b1.w, h1 = b2.x, cf1 = b2.y;
        float pc0 = b2.z, pc1 = b2.w;

        // ground truth
        float gtx = g0.x, gty = g0.y, gtw = g0.z, gth = g0.w;
        float gconf = g1.x;
        float glx = g1.y, guy = g1.z, grx = g1.w, gdy = g2.x;
        float exist = g2.y;
        float gc0 = g2.z, gc1 = g2.w;

        float g_area = (grx - glx) * (gdy - guy);

        float iou0 = iou_box(x0, y0, w0, h0, gxc, gyc, glx, guy, grx, gdy, g_area);
        float iou1 = iou_box(x1, y1, w1, h1, gxc, gyc, glx, guy, grx, gdy, g_area);

        // argmax picks the FIRST max on ties -> box 1 only if strictly greater
        bool sel1 = iou1 > iou0;
        float max_iou = sel1 ? iou1 : iou0;
        float m = ((exist != 0.0f) && (max_iou > 0.0f)) ? 1.0f : 0.0f;

        float pmx = sel1 ? x1 : x0;
        float pmy = sel1 ? y1 : y0;
        float pmw = sel1 ? w1 : w0;
        float pmh = sel1 ? h1 : h0;
        float pmc = sel1 ? cf1 : cf0;

        float dx = gtx - pmx;
        float dy = gty - pmy;
        float dw = sqrtf(gtw) - sqrtf(pmw);
        float dh = sqrtf(gth) - sqrtf(pmh);
        float coord = L_COORD * (dx * dx + dy * dy + dw * dw + dh * dh);

        float dc = gconf - pmc;
        float conf_obj = dc * dc;

        float e0 = gc0 - pc0;
        float e1 = gc1 - pc1;
        float cls = L_NOOBJ * (e0 * e0 + e1 * e1);

        float conf_noobj = L_NOOBJ * (cf0 * cf0 + cf1 * cf1 - m * pmc * pmc);

        acc += m * (coord + conf_obj + cls) + conf_noobj;
    }

    // ---- Wave32 reduction through the matrix pipe ----
    // D = ones(16x4) x B + 0, with B holding the 32 per-lane partials in its
    // first VGPR (second VGPR zero). D[m][n] = sum_k B[k][n]; row M=0 of D
    // (lanes 0..15, VGPR0) holds the 16 column sums whose total equals the
    // sum of all 32 lane partials, exactly in f32. All lanes active here
    // (full 256-thread blocks, loop exited uniformly) -> EXEC all ones.
    v2f a; a[0] = 1.0f; a[1] = 1.0f;
    v2f b; b[0] = acc;  b[1] = 0.0f;
    v8f cz = {};
    v8f d = __builtin_amdgcn_wmma_f32_16x16x4_f32(
        /*neg_a=*/false, a, /*neg_b=*/false, b,
        /*c_mod=*/(short)0, cz, /*reuse_a=*/false, /*reuse_b=*/false);

    float wsum = d[0];
    wsum += __shfl_xor(wsum, 1, 32);
    wsum += __shfl_xor(wsum, 2, 32);
    wsum += __shfl_xor(wsum, 4, 32);
    wsum += __shfl_xor(wsum, 8, 32);
    // lanes 0..15 and 16..31 each now hold the full wave total

    __shared__ float smem[8];           // 8 waves per 256-thread block (wave32)
    int lane = threadIdx.x & 31;
    int wave = threadIdx.x >> 5;
    if (lane == 0) smem[wave] = wsum;
    __syncthreads();

    if (wave == 0) {
        float v = (lane < 8) ? smem[lane] : 0.0f;
        v += __shfl_xor(v, 1, 32);
        v += __shfl_xor(v, 2, 32);
        v += __shfl_xor(v, 4, 32);
        if (lane == 0) atomicAdd(out, v);
    }
}

extern "C" void kernel_launch(void* const* d_in, const int* in_sizes, int n_in,
                              void* d_out, int out_size, void* d_ws, size_t ws_size,
                              hipStream_t stream) {
    const float* bb = (const float*)d_in[0];   // bounding_boxes (N,7,7,12) f32
    const float* gt = (const float*)d_in[1];   // ground_truth  (N,7,7,1,12) f32
    float* out = (float*)d_out;

    int cells = in_sizes[0] / YCH;             // N * 7 * 7

    yolo_zero_kernel<<<1, 64, 0, stream>>>(out);

    int threads = 256;
    int blocks = 2048;                         // ~16K wave32 waves: saturates HBM
    int needed = (cells + threads - 1) / threads;
    if (blocks > needed) blocks = needed;
    if (blocks < 1) blocks = 1;

    yolo_loss_kernel<<<blocks, threads, 0, stream>>>(bb, gt, out, cells);
}